// GCNModel_8589934592403
// MI455X (gfx1250) — compile-verified
//
#include <hip/hip_runtime.h>

typedef __attribute__((ext_vector_type(2))) float v2f;
typedef __attribute__((ext_vector_type(8))) float v8f;

#define D 64   // feature width of both GCN layers
#define DOUT 8

// ---------------- degree / normalization ----------------

__global__ void k_init_deg(float* __restrict__ deg, int n) {
  int i = blockIdx.x * blockDim.x + threadIdx.x;
  if (i < n) deg[i] = 1.0f;                      // self-loop contributes 1
}

__global__ void k_count_deg(const long long* __restrict__ dst,
                            float* __restrict__ deg, int ne) {
  int e = blockIdx.x * blockDim.x + threadIdx.x;
  if (e < ne) {
    __builtin_prefetch(&dst[e + 8192], 0, 0);    // stream edge list from HBM
    atomicAdd(&deg[(int)dst[e]], 1.0f);
  }
}

__global__ void k_dinv(float* __restrict__ deg, int n) {
  int i = blockIdx.x * blockDim.x + threadIdx.x;
  if (i < n) {
    float d = deg[i];
    deg[i] = (d > 0.0f) ? rsqrtf(d) : 0.0f;      // in place: deg -> dinv
  }
}

__global__ void k_zero4(float4* __restrict__ p, long n4) {
  long i = (long)blockIdx.x * blockDim.x + threadIdx.x;
  if (i < n4) p[i] = make_float4(0.f, 0.f, 0.f, 0.f);
}

// ---------------- dense GEMM: H[nrows x 64] = X[nrows x 64] @ W[64 x 64] ----
// One wave produces a 16-row slab across all 64 output columns using
// V_WMMA_F32_16X16X4_F32 (16 K-steps x 4 N-tiles = 64 WMMA per wave).
// W is staged in LDS pair-interleaved: Wp[p*64 + col] = {W[2p][col], W[2p+1][col]}
// so every B-fragment is ONE aligned ds_load_b64 directly into the even VGPR
// pair the WMMA wants (no v_mov shuffle chains).

__global__ void k_gemm64(const float* __restrict__ X, const float* __restrict__ W,
                         float* __restrict__ H, int nrows) {
  __shared__ v2f Wp[(D / 2) * D];                // 32 K-pairs x 64 cols, 16 KB
  for (int i = threadIdx.x; i < D * D; i += blockDim.x) {
    int k = i >> 6, col = i & 63;                // W row-major [k][col]
    ((float*)&Wp[(k >> 1) * D + col])[k & 1] = W[i];
  }
  __syncthreads();

  const int lane = threadIdx.x & 31;
  const int wv   = threadIdx.x >> 5;
  const int tile = blockIdx.x * (blockDim.x >> 5) + wv;
  const int ntiles = (nrows + 15) >> 4;
  if (tile >= ntiles) return;

  const long row0 = (long)tile * 16;
  const int  mrow = lane & 15;                   // A: row within slab
  const int  half = lane >> 4;                   // 0: K+0/1, 1: K+2/3
  const int  col  = lane & 15;                   // B/D: column within 16-tile

  long arow = row0 + mrow;
  if (arow >= nrows) arow = nrows - 1;           // clamp (stores are guarded)
  const float* xrow = X + arow * (long)D;

  v8f acc0 = {}, acc1 = {}, acc2 = {}, acc3 = {};

#pragma unroll
  for (int kk = 0; kk < D / 4; ++kk) {           // K advances 4 per step
    const v2f a = *(const v2f*)(xrow + kk * 4 + half * 2);   // global_load_b64

    const v2f* wrow = &Wp[(kk * 2 + half) * D + col];
    const v2f b0 = wrow[0];                      // ds_load_b64 each
    const v2f b1 = wrow[16];
    const v2f b2 = wrow[32];
    const v2f b3 = wrow[48];

    acc0 = __builtin_amdgcn_wmma_f32_16x16x4_f32(false, a, false, b0, (short)0, acc0, false, false);
    acc1 = __builtin_amdgcn_wmma_f32_16x16x4_f32(false, a, false, b1, (short)0, acc1, false, false);
    acc2 = __builtin_amdgcn_wmma_f32_16x16x4_f32(false, a, false, b2, (short)0, acc2, false, false);
    acc3 = __builtin_amdgcn_wmma_f32_16x16x4_f32(false, a, false, b3, (short)0, acc3, false, false);
  }

  // D-matrix layout: VGPR r -> row r (lanes 0-15) / row r+8 (lanes 16-31)
  const int rb = half << 3;                      // 0 or 8
#pragma unroll
  for (int r = 0; r < 8; ++r) {
    long row = row0 + rb + r;
    if (row < nrows) {
      float* dp = H + row * (long)D + col;
      dp[0]  = acc0[r];
      dp[16] = acc1[r];
      dp[32] = acc2[r];
      dp[48] = acc3[r];
    }
  }
}

// ---------------- edge gather + scale + scatter-add ----------------
// One wave per edge: 32 lanes x float2 cover the 64 features. Gather from H
// (L2-resident, 25.6 MB), scatter with global f32 atomics to AGG.

__global__ void k_edge(const long long* __restrict__ src,
                       const long long* __restrict__ dst,
                       const float* __restrict__ dinv,
                       const float* __restrict__ H,
                       float* __restrict__ AGG, int ne) {
  const int lane = threadIdx.x & 31;
  const int wv   = threadIdx.x >> 5;
  const long e = (long)blockIdx.x * (blockDim.x >> 5) + wv;
  if (e >= ne) return;

  if (lane == 0) {                               // stream edge list from HBM
    __builtin_prefetch(&src[e + 4096], 0, 0);
    __builtin_prefetch(&dst[e + 4096], 0, 0);
  }

  const int s = (int)src[e];
  const int d = (int)dst[e];
  const float norm = dinv[s] * dinv[d];

  const float2 hv = ((const float2*)(H + (long)s * D))[lane];
  float* ap = AGG + (long)d * D + lane * 2;
  atomicAdd(ap,     norm * hv.x);
  atomicAdd(ap + 1, norm * hv.y);
}

// ---------------- self-loop + bias + ReLU (in place on AGG) ----------------

__global__ void k_finalize(float* __restrict__ AGG, const float* __restrict__ H,
                           const float* __restrict__ dinv,
                           const float* __restrict__ bias, int n) {
  const long i = (long)blockIdx.x * blockDim.x + threadIdx.x;   // per float4
  const long total = (long)n * (D / 4);
  if (i >= total) return;
  const int node = (int)(i >> 4);
  const int f4   = (int)(i & 15);

  const float di = dinv[node];
  const float s  = di * di;                      // self-loop norm
  const float4 hv = ((const float4*)H)[i];
  float4 av       = ((float4*)AGG)[i];
  const float4 bv = ((const float4*)bias)[f4];

  av.x = fmaxf(av.x + s * hv.x + bv.x, 0.f);
  av.y = fmaxf(av.y + s * hv.y + bv.y, 0.f);
  av.z = fmaxf(av.z + s * hv.z + bv.z, 0.f);
  av.w = fmaxf(av.w + s * hv.w + bv.w, 0.f);
  ((float4*)AGG)[i] = av;
}

// ---------------- output projection: OUT[n x 8] = H[n x 64] @ Wout + bout ---
// One wave per node; lane holds 2 features, 8 partial sums, wave reduction.

__global__ void k_out(const float* __restrict__ H, const float* __restrict__ Wout,
                      const float* __restrict__ bout, float* __restrict__ OUT, int n) {
  __shared__ float Ws[D * DOUT];
  for (int i = threadIdx.x; i < D * DOUT; i += blockDim.x) Ws[i] = Wout[i];
  __syncthreads();

  const int lane = threadIdx.x & 31;
  const int wv   = threadIdx.x >> 5;
  const long node = (long)blockIdx.x * (blockDim.x >> 5) + wv;
  if (node >= n) return;

  const float2 hv = ((const float2*)(H + node * D))[lane];
  float acc[DOUT];
#pragma unroll
  for (int o = 0; o < DOUT; ++o)
    acc[o] = hv.x * Ws[(lane * 2) * DOUT + o] + hv.y * Ws[(lane * 2 + 1) * DOUT + o];

#pragma unroll
  for (int off = 16; off >= 1; off >>= 1)
#pragma unroll
    for (int o = 0; o < DOUT; ++o)
      acc[o] += __shfl_down(acc[o], off, 32);

  if (lane == 0) {
#pragma unroll
    for (int o = 0; o < DOUT; ++o)
      OUT[node * DOUT + o] = acc[o] + bout[o];
  }
}

// ---------------- launch ----------------

extern "C" void kernel_launch(void* const* d_in, const int* in_sizes, int n_in,
                              void* d_out, int out_size, void* d_ws, size_t ws_size,
                              hipStream_t stream) {
  const float*     x    = (const float*)d_in[0];
  const long long* ei   = (const long long*)d_in[1];  // [2, E] int64
  const float*     W1   = (const float*)d_in[2];
  const float*     b1   = (const float*)d_in[3];
  const float*     W2   = (const float*)d_in[4];
  const float*     b2   = (const float*)d_in[5];
  const float*     Wout = (const float*)d_in[6];
  const float*     bout = (const float*)d_in[7];

  const int N = in_sizes[0] / D;
  const int E = in_sizes[1] / 2;
  const long long* srcp = ei;
  const long long* dstp = ei + E;

  // workspace carve-out: dinv[N], H[N*64], AGG[N*64]  (~52 MB)
  char* ws = (char*)d_ws;
  size_t off = 0;
  float* dinv = (float*)(ws + off);
  off += (((size_t)N * sizeof(float)) + 255) & ~(size_t)255;
  float* Hbuf = (float*)(ws + off);
  off += (((size_t)N * D * sizeof(float)) + 255) & ~(size_t)255;
  float* Agg  = (float*)(ws + off);

  const int ntiles = (N + 15) / 16;
  const dim3 gemmGrid((ntiles + 3) / 4);          // 4 waves/block, 1 slab/wave
  const dim3 gemmBlk(128);
  const long n4 = (long)N * (D / 4);
  const int zeroBlocks = (int)((n4 + 255) / 256);
  const int edgeBlocks = (E + 7) / 8;             // 8 waves/block, 1 edge/wave
  const int finBlocks  = zeroBlocks;
  const int nodeBlocks = (N + 255) / 256;
  const int outBlocks  = (N + 7) / 8;

  // normalization
  k_init_deg <<<nodeBlocks, 256, 0, stream>>>(dinv, N);
  k_count_deg<<<(E + 255) / 256, 256, 0, stream>>>(dstp, dinv, E);
  k_dinv     <<<nodeBlocks, 256, 0, stream>>>(dinv, N);

  // layer 1
  k_gemm64  <<<gemmGrid, gemmBlk, 0, stream>>>(x, W1, Hbuf, N);
  k_zero4   <<<zeroBlocks, 256, 0, stream>>>((float4*)Agg, n4);
  k_edge    <<<edgeBlocks, 256, 0, stream>>>(srcp, dstp, dinv, Hbuf, Agg, E);
  k_finalize<<<finBlocks, 256, 0, stream>>>(Agg, Hbuf, dinv, b1, N);

  // layer 2 (Agg holds activated layer-1 output; GEMM consumes it, then reuse)
  k_gemm64  <<<gemmGrid, gemmBlk, 0, stream>>>(Agg, W2, Hbuf, N);
  k_zero4   <<<zeroBlocks, 256, 0, stream>>>((float4*)Agg, n4);
  k_edge    <<<edgeBlocks, 256, 0, stream>>>(srcp, dstp, dinv, Hbuf, Agg, E);
  k_finalize<<<finBlocks, 256, 0, stream>>>(Agg, Hbuf, dinv, b2, N);

  // output projection
  k_out<<<outBlocks, 256, 0, stream>>>(Agg, Wout, bout, (float*)d_out, N);
}